// NoiseModel_64089501991325
// MI455X (gfx1250) — compile-verified
//
#include <hip/hip_runtime.h>
#include <hip/hip_bf16.h>
#include <stdint.h>

// ---------------------------------------------------------------------------
// NoiseModel: out = PoissonSample(image*4095*ksi/16) * (16/4095) + noise_pattern
// Memory-bound streaming kernel (~302 MB / call -> ~13us floor @ 23.3 TB/s).
// CDNA5 async data mover (global_load_async_to_lds_b128 + ASYNCcnt) streams the
// image through LDS with a 2-deep software pipeline; noise_pattern streams
// directly via global_load_b128 as a second concurrent HBM stream.
// ---------------------------------------------------------------------------

#define TPB 256
#define CHUNK_FLOATS 1024u /* TPB lanes * 4 floats (one b128 per lane) */

__device__ __forceinline__ unsigned pcg32(unsigned v) {
    v = v * 747796405u + 2891336453u;
    unsigned w = ((v >> ((v >> 28u) + 4u)) ^ v) * 277803737u;
    return (w >> 22u) ^ w;
}

__device__ __forceinline__ float u01(unsigned r) {
    // (0,1) uniform from top 24 bits
    return (float)(r >> 8) * 0x1.0p-24f + 0x1.0p-25f;
}

__device__ __forceinline__ float poisson_sample(float rate, unsigned idx) {
    // Counter-based RNG: deterministic, stateless (seed 42 folded into constants).
    unsigned h  = idx * 0x9E3779B9u + 0x2545F491u;
    unsigned r1 = pcg32(h);
    unsigned r2 = pcg32(h ^ 0x68BC21EBu);
    float s;
    if (rate < 12.0f) {
        // Knuth product-of-uniforms; expected iterations ~ rate+1 (< 13)
        float L = __expf(-rate);
        float p = 1.0f;
        float k = -1.0f;
        unsigned st = r1;
        do {
            k += 1.0f;
            st = pcg32(st);
            p *= u01(st);
        } while (p > L && k < 40.0f);
        s = k;
    } else {
        // Normal approximation with continuity correction (rate up to ~256)
        float z = __fsqrt_rn(-2.0f * __logf(u01(r1))) * __cosf(6.28318530718f * u01(r2));
        s = floorf(__fmaf_rn(__fsqrt_rn(rate), z, rate) + 0.5f);
    }
    return fmaxf(s, 0.0f);
}

__global__ void __launch_bounds__(TPB)
noise_model_kernel(const float* __restrict__ img,
                   const float* __restrict__ ksi,
                   const float* __restrict__ npat,
                   float* __restrict__ out,
                   unsigned n, unsigned nfull, unsigned per_b) {
    __shared__ float smem[2][CHUNK_FLOATS];
    const unsigned tid = threadIdx.x;
    // Generic->LDS aperture pointers: low 32 bits are the LDS byte offset.
    const unsigned lds0 = (unsigned)(size_t)&smem[0][tid * 4];
    const unsigned lds1 = (unsigned)(size_t)&smem[1][tid * 4];
    const unsigned stride = gridDim.x;
    const float R2P = 4095.0f / 16.0f;  // rate = image * SCALE * ksi / GAIN
    const float P2O = 16.0f / 4095.0f;  // out  = sample * GAIN / SCALE + noise

    unsigned chunk = blockIdx.x;
    unsigned buf = 0;

    // Prime the pipeline: async DMA of first chunk of image into LDS buffer 0.
    if (chunk < nfull) {
        unsigned goff = chunk * 4096u + tid * 16u;
        asm volatile("global_load_async_to_lds_b128 %0, %1, %2"
                     :: "v"(lds0), "v"(goff), "s"(img) : "memory");
    }

    for (; chunk < nfull; chunk += stride) {
        const unsigned next = chunk + stride;
        if (next < nfull) {
            // Issue next chunk into the other buffer, then wait for the
            // older transfer only (async loads retire in order).
            unsigned goff = next * 4096u + tid * 16u;
            asm volatile("global_load_async_to_lds_b128 %0, %1, %2"
                         :: "v"(buf ? lds0 : lds1), "v"(goff), "s"(img) : "memory");
            asm volatile("s_wait_asynccnt 1" ::: "memory");
        } else {
            asm volatile("s_wait_asynccnt 0" ::: "memory");
        }

        const unsigned i0 = chunk * CHUNK_FLOATS + tid * 4u;
        const float4 im = *reinterpret_cast<const float4*>(&smem[buf][tid * 4]);
        const float4 nz = *reinterpret_cast<const float4*>(npat + i0);
        // Batch index is uniform across the block (per_b is a multiple of the
        // chunk size) -> force scalar so ksi comes via s_load (constant cache).
        const int b = __builtin_amdgcn_readfirstlane((int)(i0 / per_b));
        const float coef = ksi[b] * R2P;

        float4 o;
        o.x = __fmaf_rn(poisson_sample(im.x * coef, i0 + 0u), P2O, nz.x);
        o.y = __fmaf_rn(poisson_sample(im.y * coef, i0 + 1u), P2O, nz.y);
        o.z = __fmaf_rn(poisson_sample(im.z * coef, i0 + 2u), P2O, nz.z);
        o.w = __fmaf_rn(poisson_sample(im.w * coef, i0 + 3u), P2O, nz.w);
        *reinterpret_cast<float4*>(out + i0) = o;

        buf ^= 1u;
    }

    // Scalar tail (n not a multiple of CHUNK_FLOATS; unused for 8x3x1024x1024)
    const unsigned tail0 = nfull * CHUNK_FLOATS;
    for (unsigned i = tail0 + blockIdx.x * TPB + tid; i < n; i += stride * TPB) {
        const float coef = ksi[i / per_b] * R2P;
        out[i] = __fmaf_rn(poisson_sample(img[i] * coef, i), P2O, npat[i]);
    }
}

extern "C" void kernel_launch(void* const* d_in, const int* in_sizes, int n_in,
                              void* d_out, int out_size, void* d_ws, size_t ws_size,
                              hipStream_t stream) {
    (void)d_ws; (void)ws_size;
    const float* img  = (const float*)d_in[0];
    const float* ksi  = (const float*)d_in[1];
    const float* npat = (const float*)d_in[2];
    float* out = (float*)d_out;

    if (out_size <= 0) return;
    unsigned n = (unsigned)out_size;

    unsigned nb = (n_in > 1 && in_sizes[1] > 0) ? (unsigned)in_sizes[1] : 1u;
    unsigned per_b = n / nb;
    if (per_b == 0) per_b = n;

    unsigned nfull = n / CHUNK_FLOATS;
    unsigned grid = 4096;            // grid-stride: ~6 chunks/block for this size
    if (nfull > 0 && nfull < grid) grid = nfull;
    if (nfull == 0) grid = (n + TPB - 1) / TPB;
    if (grid < 1) grid = 1;

    noise_model_kernel<<<dim3(grid), dim3(TPB), 0, stream>>>(img, ksi, npat, out,
                                                             n, nfull, per_b);
}